// YunetPostProcessing_3212635538202
// MI455X (gfx1250) — compile-verified
//
#include <hip/hip_runtime.h>

// YuNet post-processing for MI455X (gfx1250, wave32).
// decode (grid) -> TDM tensor_load_to_lds (keys, 64KB) + async global->LDS
// (bboxes, 128KB) -> bitonic sort in LDS -> wave-0 greedy NMS (LDS-resident
// boxes, ballot_w32) -> gather 50x15 dets.

#define N_BOXES      8192
#define TOP_K        50
#define IOU_THR      0.3f
#define SORT_THREADS 1024

typedef __attribute__((ext_vector_type(4))) unsigned u32x4;
typedef __attribute__((ext_vector_type(8))) unsigned u32x8;

// ---------------------------------------------------------------------------
// Kernel 1: decode boxes / landmarks / scores, build sort keys.
// keys[i] = (float_bits(score) << 32) | i   (score >= 0 so bit order == value order)
// ---------------------------------------------------------------------------
__global__ __launch_bounds__(256) void yunet_decode_kernel(
    const float* __restrict__ loc, const float* __restrict__ conf,
    const float* __restrict__ iou, const float* __restrict__ priors,
    unsigned long long* __restrict__ keys,
    float4* __restrict__ bbox, float* __restrict__ lms)
{
    int i = blockIdx.x * blockDim.x + threadIdx.x;
    if (i >= N_BOXES) return;

    // gfx1250 prefetch path (global_prefetch_b8)
    if (i + 4096 < N_BOXES) __builtin_prefetch(&loc[(i + 4096) * 14], 0, 1);

    float l0 = loc[i * 14 + 0], l1 = loc[i * 14 + 1];
    float l2 = loc[i * 14 + 2], l3 = loc[i * 14 + 3];

    float cls = conf[i * 2 + 1];
    float q   = fminf(fmaxf(iou[i], 0.0f), 1.0f);
    float score = sqrtf(cls * q);

    float cx = priors[i * 4 + 0], cy = priors[i * 4 + 1];
    float w  = priors[i * 4 + 2], h  = priors[i * 4 + 3];

    float bcx = fmaf(l0 * 0.1f, w, cx);
    float bcy = fmaf(l1 * 0.1f, h, cy);
    float hw  = w * expf(l2 * 0.1f) * 0.5f;   // var = [0.1, 0.2]
    float hh  = h * expf(l3 * 0.2f) * 0.5f;

    bbox[i] = make_float4(bcx - hw, bcy - hh, bcx + hw, bcy + hh);

#pragma unroll
    for (int k = 0; k < 5; ++k) {
        float lx = loc[i * 14 + 4 + 2 * k];
        float ly = loc[i * 14 + 5 + 2 * k];
        lms[i * 10 + 2 * k + 0] = fmaf(lx * 0.1f, w, cx);  // landmarks use VAR0 both axes
        lms[i * 10 + 2 * k + 1] = fmaf(ly * 0.1f, h, cy);
    }

    keys[i] = ((unsigned long long)__float_as_uint(score) << 32) | (unsigned)i;
}

// ---------------------------------------------------------------------------
// Kernel 2: single workgroup (192KB LDS resident working set).
// ---------------------------------------------------------------------------
__global__ __launch_bounds__(SORT_THREADS) void yunet_sort_nms_kernel(
    const unsigned long long* __restrict__ keys,
    const float4* __restrict__ bbox, const float* __restrict__ lms,
    float* __restrict__ out)
{
    __shared__ __align__(16) unsigned long long skeys[N_BOXES];  // 64 KB
    __shared__ __align__(16) float sbbox[N_BOXES * 4];           // 128 KB
    const unsigned tid = threadIdx.x;

    // Low 32 bits of a generic pointer to LDS are the LDS byte offset.
    unsigned lds_keys = (unsigned)(unsigned long long)(&skeys[0]);
    unsigned lds_bbox = (unsigned)(unsigned long long)(&sbbox[0]);

    // ---- async global -> LDS staging of bboxes (ASYNCcnt path) ----
    // 128 bytes per thread = 8 x b128 per wave-lane; overlapped with the sort.
#pragma unroll
    for (int q = 0; q < 8; ++q) {
        unsigned byteoff = tid * 128u + (unsigned)q * 16u;
        asm volatile("global_load_async_to_lds_b128 %0, %1, %2"
                     :
                     : "v"(lds_bbox + byteoff), "v"(byteoff), "s"(bbox)
                     : "memory");
    }

    // ---- TDM: one descriptor moves all 64KB of keys to LDS (TENSORcnt) ----
    if (tid < 32) {   // one wave issues the DMA (TDM ignores EXEC)
        unsigned long long ga = (unsigned long long)keys;
        u32x4 g0;
        g0[0] = 1u;                                   // count=1, user-mode D#
        g0[1] = lds_keys;                             // lds_addr (bytes)
        g0[2] = (unsigned)ga;                         // global_addr[31:0]
        g0[3] = ((unsigned)(ga >> 32) & 0x01FFFFFFu)  // global_addr[56:32]
                | 0x80000000u;                        // type=2 ("image")
        u32x8 g1;
        g1[0] = 3u << 16;                             // data_size=8B, wg_mask=0
        g1[1] = (N_BOXES & 0xFFFFu) << 16;            // tensor_dim0[15:0]
        g1[2] = (N_BOXES >> 16) | (1u << 16);         // tensor_dim0[31:16] | tensor_dim1=1
        g1[3] = (unsigned)N_BOXES << 16;              // tensor_dim1[31:16]=0 | tile_dim0=8192
        g1[4] = 1u;                                   // tile_dim1=1, tile_dim2=0
        g1[5] = N_BOXES;                              // tensor_dim0_stride[31:0]
        g1[6] = (N_BOXES & 0xFFFFu) << 16;            // stride0[47:32]=0 | stride1[15:0]
        g1[7] = 0u;                                   // tensor_dim1_stride[47:16]
        asm volatile("tensor_load_to_lds %0, %1" : : "s"(g0), "s"(g1) : "memory");
    }
    __builtin_amdgcn_s_wait_tensorcnt(0);
    __syncthreads();          // keys visible to all waves; bbox copy still in flight

    // ---- bitonic sort, descending (u64 keys => descending scores) ----
    for (unsigned k = 2; k <= N_BOXES; k <<= 1) {
        for (unsigned j = k >> 1; j > 0; j >>= 1) {
            __syncthreads();
            for (unsigned p = tid; p < N_BOXES / 2; p += SORT_THREADS) {
                unsigned i = ((p & ~(j - 1u)) << 1) | (p & (j - 1u));
                unsigned long long a = skeys[i];
                unsigned long long b = skeys[i + j];
                bool desc = ((i & k) == 0u);
                if ((a < b) == desc) { skeys[i] = b; skeys[i + j] = a; }
            }
        }
    }
    // every wave drains its own async copies, then rendezvous before NMS reads
    asm volatile("s_wait_asynccnt 0" ::: "memory");
    __syncthreads();

    const float4* bb4 = (const float4*)sbbox;

    // ---- greedy NMS on wave 0: kept boxes live in registers, 2 slots/lane ----
    if (tid < 32) {
        int lane = (int)tid;
        float kx1[2] = {0.f, 0.f}, ky1[2] = {0.f, 0.f};
        float kx2[2] = {0.f, 0.f}, ky2[2] = {0.f, 0.f}, kar[2] = {0.f, 0.f};
        int kept = 0;
        for (int cand = 0; cand < N_BOXES && kept < TOP_K; ++cand) {
            unsigned long long key = skeys[cand];
            unsigned idx = (unsigned)key;
            float4 b = bb4[idx];                      // LDS-resident boxes
            float area = (b.z - b.x) * (b.w - b.y);
            bool sup = false;
#pragma unroll
            for (int s = 0; s < 2; ++s) {
                if (s * 32 + lane < kept) {
                    float iw = fmaxf(fminf(b.z, kx2[s]) - fmaxf(b.x, kx1[s]), 0.0f);
                    float ih = fmaxf(fminf(b.w, ky2[s]) - fmaxf(b.y, ky1[s]), 0.0f);
                    float inter = iw * ih;
                    float r = inter / (area + kar[s] - inter);
                    sup = sup || (r > IOU_THR);
                }
            }
            if (__builtin_amdgcn_ballot_w32(sup) == 0u) {
                if (lane == (kept & 31)) {
                    int slot = kept >> 5;
                    kx1[slot] = b.x; ky1[slot] = b.y;
                    kx2[slot] = b.z; ky2[slot] = b.w; kar[slot] = area;
                }
                // overwrite consumed prefix of the sorted list with kept keys
                if (lane == 0) skeys[kept] = key;
                ++kept;
            }
        }
        // reference fill semantics: nonzero(..., fill_value=0) -> order[0],
        // which is always the first kept box => skeys[0].
        unsigned long long k0 = skeys[0];
        for (int r = kept + lane; r < TOP_K; r += 32) skeys[r] = k0;
    }
    __syncthreads();

    // ---- gather dets: 50 rows x [x1 y1 x2 y2, 10 lms, score] ----
    for (int e = (int)tid; e < TOP_K * 15; e += SORT_THREADS) {
        int row = e / 15, col = e - row * 15;
        unsigned long long key = skeys[row];
        unsigned idx = (unsigned)key;
        float v;
        if (col < 4)       v = sbbox[idx * 4 + col];
        else if (col < 14) v = lms[idx * 10 + (col - 4)];
        else               v = __uint_as_float((unsigned)(key >> 32));
        out[e] = v;
    }
}

// ---------------------------------------------------------------------------
extern "C" void kernel_launch(void* const* d_in, const int* in_sizes, int n_in,
                              void* d_out, int out_size, void* d_ws, size_t ws_size,
                              hipStream_t stream) {
    const float* loc    = (const float*)d_in[0];
    const float* conf   = (const float*)d_in[1];
    const float* iou    = (const float*)d_in[2];
    const float* priors = (const float*)d_in[3];

    // workspace layout: keys 64KB | bbox 128KB | lms 320KB  (512KB total)
    unsigned long long* keys = (unsigned long long*)d_ws;
    float4* bbox = (float4*)((char*)d_ws + 64 * 1024);
    float*  lms  = (float*)((char*)d_ws + 64 * 1024 + 128 * 1024);

    yunet_decode_kernel<<<N_BOXES / 256, 256, 0, stream>>>(
        loc, conf, iou, priors, keys, bbox, lms);
    yunet_sort_nms_kernel<<<1, SORT_THREADS, 0, stream>>>(
        keys, bbox, lms, (float*)d_out);
}